// ColorEh_317827580559
// MI455X (gfx1250) — compile-verified
//
#include <hip/hip_runtime.h>
#include <hip/hip_bf16.h>

typedef __attribute__((ext_vector_type(2))) float v2f;
typedef __attribute__((ext_vector_type(8))) float v8f;

#define SW 58            // LDS tile row stride (even -> 8B-aligned v2f, few conflicts)
#define TILE_F (16 * SW) // floats per 16-row tile buffer

struct KP {
    const float* fea;
    const long long* link;
    const float* invn;   // 3 floats in ws
    float* fea2;         // N x 54 in ws
    float* out;          // N x 102
    int n;
    // padded weights/biases in ws: W rows padded to NP=ceil16(NOUT),
    // cols to KPAD=ceil4(KIN), zero-filled; bias padded to NP.
    const float* w11[3]; const float* b11[3];
    const float* w21[3]; const float* b21[3];
    const float* w31[3]; const float* b31[3];
    const float* w41[3]; const float* b41[3];
    const float* w22[3]; const float* b22[3];
    const float* w23[3]; const float* b23[3]; // w23[0]: 64 x 504 (9 slices of 56)
};

// ---------------- weight padding prep ---------------------------------------
struct PD { const float* src; int dst, kin, nout, kpad, np, mode; };
struct PA { PD d[36]; float* base; };

__global__ __launch_bounds__(256) void k_prep(PA a) {
    PD d = a.d[blockIdx.x];
    float* dst = a.base + d.dst;
    const int total = d.np * d.kpad;
    for (int e = threadIdx.x; e < total; e += 256) {
        int n = e / d.kpad, k = e % d.kpad;
        float v = 0.f;
        if (d.mode == 0) {
            if (n < d.nout && k < d.kin) v = d.src[n * d.kin + k];
        } else { // fc23 W0: 54x486 -> 64x504 as 9 slices of stride 56
            int j = k / 56, kk = k % 56;
            if (n < 54 && kk < 54) v = d.src[n * 486 + j * 54 + kk];
        }
        dst[e] = v;
    }
}

// ---------------- deterministic column sum-of-squares (cols 0..2) ----------
__global__ __launch_bounds__(256) void k_norm1(const float* __restrict__ fea,
                                               float* __restrict__ partial, int n) {
    __shared__ float s[3][256];
    int tid = threadIdx.x;
    float a0 = 0.f, a1 = 0.f, a2 = 0.f;
    for (int i = blockIdx.x * 256 + tid; i < n; i += 512 * 256) {
        float f0 = fea[i * 9 + 0], f1 = fea[i * 9 + 1], f2 = fea[i * 9 + 2];
        a0 += f0 * f0; a1 += f1 * f1; a2 += f2 * f2;
    }
    s[0][tid] = a0; s[1][tid] = a1; s[2][tid] = a2;
    __syncthreads();
    for (int off = 128; off > 0; off >>= 1) {
        if (tid < off) {
            s[0][tid] += s[0][tid + off];
            s[1][tid] += s[1][tid + off];
            s[2][tid] += s[2][tid + off];
        }
        __syncthreads();
    }
    if (tid == 0) {
        partial[blockIdx.x * 3 + 0] = s[0][0];
        partial[blockIdx.x * 3 + 1] = s[1][0];
        partial[blockIdx.x * 3 + 2] = s[2][0];
    }
}

__global__ __launch_bounds__(256) void k_norm2(const float* __restrict__ partial,
                                               float* __restrict__ invn) {
    __shared__ float s[3][256];
    int tid = threadIdx.x;
    float a0 = 0.f, a1 = 0.f, a2 = 0.f;
    for (int b = tid; b < 512; b += 256) {
        a0 += partial[b * 3 + 0]; a1 += partial[b * 3 + 1]; a2 += partial[b * 3 + 2];
    }
    s[0][tid] = a0; s[1][tid] = a1; s[2][tid] = a2;
    __syncthreads();
    for (int off = 128; off > 0; off >>= 1) {
        if (tid < off) {
            s[0][tid] += s[0][tid + off];
            s[1][tid] += s[1][tid + off];
            s[2][tid] += s[2][tid + off];
        }
        __syncthreads();
    }
    if (tid < 3) invn[tid] = 1.f / fmaxf(sqrtf(s[tid][0]), 1e-12f);
}

// ---------------- wave-level GEMM: y(16xNOUT) = x(16xKIN) @ W^T + b --------
// Padded weights => no guards: rows n>=NOUT are zero (acc stays 0), cols
// k>=KIN are zero.  All A/B fragment loads are aligned 8-byte pairs.
template <int KIN, int NOUT, bool RELU>
__device__ __forceinline__ void wave_gemm(const float* __restrict__ xb,
                                          float* __restrict__ yb,
                                          const float* __restrict__ Wp,
                                          const float* __restrict__ Bp, int lane) {
    const int half = lane >> 4;   // 0: K={k,k+1}, 1: K={k+2,k+3}
    const int l15 = lane & 15;
    constexpr int NT = (NOUT + 15) / 16;
    constexpr int KPAD = ((KIN + 3) / 4) * 4;
#pragma unroll
    for (int nt = 0; nt < NT; ++nt) {
        const int n = nt * 16 + l15;
        const float bn = Bp[n];
        v8f acc = {bn, bn, bn, bn, bn, bn, bn, bn};
#pragma unroll
        for (int ks = 0; ks < KPAD / 4; ++ks) {
            const int k = ks * 4 + half * 2;
            v2f a = *(const v2f*)(xb + l15 * SW + k);
            v2f bv = *(const v2f*)(Wp + n * KPAD + k);
            acc = __builtin_amdgcn_wmma_f32_16x16x4_f32(false, a, false, bv,
                                                        (short)0, acc, false, false);
        }
        const int col = nt * 16 + l15;
        if (col < 56) {
#pragma unroll
            for (int r = 0; r < 8; ++r) {
                float v = acc[r];
                if (RELU) v = fmaxf(v, 0.f);
                yb[(r + half * 8) * SW + col] = v;
            }
        }
    }
}

// copy WID floats (even) per row from LDS tile to global rows, v2f-wide
template <int WID>
__device__ __forceinline__ void tile_store(const float* __restrict__ buf,
                                           float* __restrict__ g, int gstride,
                                           int rowbase, int n, int lane) {
    constexpr int NV = WID / 2;
    for (int t = lane; t < 16 * NV; t += 32) {
        int m = t / NV, c2 = t % NV, row = rowbase + m;
        if (row < n) *(v2f*)(g + row * gstride + 2 * c2) = *(const v2f*)(buf + m * SW + 2 * c2);
    }
}

// ---------------- main MLP chain: fea6 -> fc11 -> fc21 -> fc31 -> fc41 -----
__global__ __launch_bounds__(128) void k_main(KP p) {
    const int lane = threadIdx.x & 31;
    const int wid = threadIdx.x >> 5;
    const int rowbase = blockIdx.x * 64 + wid * 16;
    __shared__ float smem[4 * 2 * TILE_F];
    float* A = &smem[wid * 2 * TILE_F];
    float* Bf = A + TILE_F;

    const float inv0 = p.invn[0], inv1 = p.invn[1], inv2 = p.invn[2];
    // build fea6 tile (cols 0..5), then zero-pad cols 6..7
    for (int t = lane; t < 16 * 6; t += 32) {
        int m = t / 6, c = t % 6, row = rowbase + m;
        float v = 0.f;
        if (row < p.n) {
            float f = p.fea[row * 9 + c];
            v = (c == 0) ? f * inv0 : (c == 1) ? f * inv1 : (c == 2) ? f * inv2
                                                                    : f * (1.0f / 255.0f);
        }
        A[m * SW + c] = v;
    }
    for (int t = lane; t < 16; t += 32) *(v2f*)(A + t * SW + 6) = (v2f)(0.f);
    __syncthreads();
    tile_store<6>(A, p.out + 96, 102, rowbase, p.n, lane); // fea6
    // fc11: 6->18->18->18, relu at end
    wave_gemm<6, 18, false>(A, Bf, p.w11[0], p.b11[0], lane); __syncthreads();
    wave_gemm<18, 18, false>(Bf, A, p.w11[1], p.b11[1], lane); __syncthreads();
    wave_gemm<18, 18, true>(A, Bf, p.w11[2], p.b11[2], lane); __syncthreads(); // fea1 in Bf
    tile_store<18>(Bf, p.out + 78, 102, rowbase, p.n, lane);
    // fc21: 18->54->54->54, relu at end
    wave_gemm<18, 54, false>(Bf, A, p.w21[0], p.b21[0], lane); __syncthreads();
    wave_gemm<54, 54, false>(A, Bf, p.w21[1], p.b21[1], lane); __syncthreads();
    wave_gemm<54, 54, true>(Bf, A, p.w21[2], p.b21[2], lane); __syncthreads(); // fea2 in A
    tile_store<54>(A, p.fea2, 54, rowbase, p.n, lane);
    // fc31: 54->18->18->18, relu at end
    wave_gemm<54, 18, false>(A, Bf, p.w31[0], p.b31[0], lane); __syncthreads();
    wave_gemm<18, 18, false>(Bf, A, p.w31[1], p.b31[1], lane); __syncthreads();
    wave_gemm<18, 18, true>(A, Bf, p.w31[2], p.b31[2], lane); __syncthreads(); // fea3 in Bf
    tile_store<18>(Bf, p.out + 6, 102, rowbase, p.n, lane);
    // fc41: 18->6->6->6, relu at end
    wave_gemm<18, 6, false>(Bf, A, p.w41[0], p.b41[0], lane); __syncthreads();
    wave_gemm<6, 6, false>(A, Bf, p.w41[1], p.b41[1], lane); __syncthreads();
    wave_gemm<6, 6, true>(Bf, A, p.w41[2], p.b41[2], lane); __syncthreads(); // fea4 in A
    tile_store<6>(A, p.out + 0, 102, rowbase, p.n, lane);
}

// ---------------- neighbor path: fc22 per neighbor, gather*mul, fc23 -------
__global__ __launch_bounds__(128) void k_nei(KP p) {
    const int lane = threadIdx.x & 31;
    const int wid = threadIdx.x >> 5;
    const int rowbase = blockIdx.x * 64 + wid * 16;
    __shared__ float sx[4][TILE_F], sy[4][TILE_F], sg[4][TILE_F];
    __shared__ float sfe[4][128];
    __shared__ int sl[4][144];
    float* X = sx[wid];
    float* Y = sy[wid];
    float* G = sg[wid];
    float* FI = sfe[wid];
    int* L = sl[wid];

    // link (0 -> self), feai (raw cols {0,1,2,6,7,8})
    for (int t = lane; t < 144; t += 32) {
        int m = t / 9, j = t % 9, row = rowbase + m;
        long long l = 0;
        if (row < p.n) {
            l = p.link[(long long)row * 9 + j];
            if (l == 0) l = row;
        }
        L[t] = (int)l;
    }
    for (int t = lane; t < 128; t += 32) {
        int m = t >> 3, c = t & 7, row = rowbase + m;
        float v = 0.f;
        if (c < 6 && row < p.n) {
            int cc = (c < 3) ? c : c + 3;
            v = p.fea[row * 9 + cc];
        }
        FI[t] = v;
    }
    __syncthreads();

    const int l15 = lane & 15, half = lane >> 4;
    v8f acc1[4]; // fc23 layer-1 accumulators, init with padded bias
#pragma unroll
    for (int nt = 0; nt < 4; ++nt) {
        float bn = p.b23[0][nt * 16 + l15];
        v8f t = {bn, bn, bn, bn, bn, bn, bn, bn};
        acc1[nt] = t;
    }

    for (int j = 0; j < 9; ++j) {
        // ninei tile for neighbor j (cols 0..5, pad 6..7)
        for (int t = lane; t < 96; t += 32) {
            int m = t / 6, c = t % 6;
            int cc = (c < 3) ? c : c + 3;
            int l = L[m * 9 + j];
            X[m * SW + c] = p.fea[l * 9 + cc] - FI[m * 8 + c];
        }
        for (int t = lane; t < 16; t += 32) *(v2f*)(X + t * SW + 6) = (v2f)(0.f);
        __syncthreads();
        // fc22: 6->54->54->54, relu at end  -> Y
        wave_gemm<6, 54, false>(X, Y, p.w22[0], p.b22[0], lane); __syncthreads();
        wave_gemm<54, 54, false>(Y, X, p.w22[1], p.b22[1], lane); __syncthreads();
        wave_gemm<54, 54, true>(X, Y, p.w22[2], p.b22[2], lane); __syncthreads();
        // gather fea2[link] rows (L2-resident), v2f-wide; pad cols 54..55
        for (int t = lane; t < 16 * 27; t += 32) {
            int m = t / 27, c2 = t % 27;
            *(v2f*)(G + m * SW + 2 * c2) = *(const v2f*)(p.fea2 + L[m * 9 + j] * 54 + 2 * c2);
        }
        for (int t = lane; t < 16; t += 32) *(v2f*)(G + t * SW + 54) = (v2f)(0.f);
        __syncthreads();
        // elementwise product -> X (cols 0..55; pads are 0*0)
        for (int t = lane; t < 16 * 28; t += 32) {
            int m = t / 28, c2 = t % 28, o = m * SW + 2 * c2;
            *(v2f*)(X + o) = *(const v2f*)(Y + o) * *(const v2f*)(G + o);
        }
        __syncthreads();
        // fc23 layer 1: accumulate slice j (padded stride 504, slice stride 56)
        const float* Wj = p.w23[0] + j * 56;
#pragma unroll
        for (int nt = 0; nt < 4; ++nt) {
            const int nn = nt * 16 + l15;
#pragma unroll
            for (int ks = 0; ks < 14; ++ks) {
                const int k = ks * 4 + half * 2;
                v2f a = *(const v2f*)(X + l15 * SW + k);
                v2f bv = *(const v2f*)(Wj + nn * 504 + k);
                acc1[nt] = __builtin_amdgcn_wmma_f32_16x16x4_f32(
                    false, a, false, bv, (short)0, acc1[nt], false, false);
            }
        }
        __syncthreads();
    }
    // spill layer-1 result to X (cols >=54 already zero via padded weights)
#pragma unroll
    for (int nt = 0; nt < 4; ++nt) {
        int col = nt * 16 + l15;
        if (col < 56) {
#pragma unroll
            for (int r = 0; r < 8; ++r) X[(r + half * 8) * SW + col] = acc1[nt][r];
        }
    }
    __syncthreads();
    wave_gemm<54, 54, false>(X, Y, p.w23[1], p.b23[1], lane); __syncthreads();
    wave_gemm<54, 54, true>(Y, X, p.w23[2], p.b23[2], lane); __syncthreads();
    tile_store<54>(X, p.out + 24, 102, rowbase, p.n, lane); // fea2_1
}

extern "C" void kernel_launch(void* const* d_in, const int* in_sizes, int n_in,
                              void* d_out, int out_size, void* d_ws, size_t ws_size,
                              hipStream_t stream) {
    KP p{};
    p.fea = (const float*)d_in[0];
    p.link = (const long long*)d_in[1];
    p.n = in_sizes[0] / 9;

    // classify MLPs by unique (b0,w0) size signature (dict ordering ambiguous)
    struct Mlp { const float* w[3]; const float* b[3]; };
    Mlp m11{}, m21{}, m31{}, m41{}, m22{}, m23{};
    int idx = 2;
    for (int g = 0; g < 6 && idx + 5 < n_in; ++g) {
        Mlp m{};
        int b0 = 0, w0 = 0;
        for (int l = 0; l < 3; ++l) {
            int sA = in_sizes[idx], sB = in_sizes[idx + 1];
            if (sA > sB) { m.w[l] = (const float*)d_in[idx]; m.b[l] = (const float*)d_in[idx + 1]; }
            else         { m.b[l] = (const float*)d_in[idx]; m.w[l] = (const float*)d_in[idx + 1]; }
            if (l == 0) { w0 = sA > sB ? sA : sB; b0 = sA > sB ? sB : sA; }
            idx += 2;
        }
        if (w0 == 108 && b0 == 18)      m11 = m;
        else if (w0 == 972 && b0 == 54) m21 = m;
        else if (w0 == 324)             m22 = m;
        else if (w0 == 26244)           m23 = m;
        else if (w0 == 972)             m31 = m;
        else                            m41 = m;
    }

    // ws layout (floats): [0..3] invn, [4..1540] partials, [2048..] padded
    // weights (<64K floats), [65536..] fea2 (N x 54)
    float* wsf = (float*)d_ws;
    p.invn = wsf;
    float* partial = wsf + 4;
    p.fea2 = wsf + 65536;
    p.out = (float*)d_out;

    PA pa{};
    pa.base = wsf;
    int nd = 0, cur = 2048;
    auto addW = [&](const float* src, int kin, int nout, int mode) -> const float* {
        int kpad = mode ? 504 : ((kin + 3) / 4) * 4;
        int np = ((nout + 15) / 16) * 16;
        pa.d[nd++] = PD{src, cur, kin, nout, kpad, np, mode};
        const float* r = wsf + cur; cur += np * kpad; return r;
    };
    auto addB = [&](const float* src, int nout) -> const float* {
        int np = ((nout + 15) / 16) * 16;
        pa.d[nd++] = PD{src, cur, 1, nout, 1, np, 0};
        const float* r = wsf + cur; cur += np; return r;
    };
    const int ki11[3] = {6, 18, 18},  no11[3] = {18, 18, 18};
    const int ki21[3] = {18, 54, 54}, no21[3] = {54, 54, 54};
    const int ki31[3] = {54, 18, 18}, no31[3] = {18, 18, 18};
    const int ki41[3] = {18, 6, 6},   no41[3] = {6, 6, 6};
    const int ki22[3] = {6, 54, 54},  no22[3] = {54, 54, 54};
    for (int l = 0; l < 3; ++l) {
        p.w11[l] = addW(m11.w[l], ki11[l], no11[l], 0); p.b11[l] = addB(m11.b[l], no11[l]);
        p.w21[l] = addW(m21.w[l], ki21[l], no21[l], 0); p.b21[l] = addB(m21.b[l], no21[l]);
        p.w31[l] = addW(m31.w[l], ki31[l], no31[l], 0); p.b31[l] = addB(m31.b[l], no31[l]);
        p.w41[l] = addW(m41.w[l], ki41[l], no41[l], 0); p.b41[l] = addB(m41.b[l], no41[l]);
        p.w22[l] = addW(m22.w[l], ki22[l], no22[l], 0); p.b22[l] = addB(m22.b[l], no22[l]);
        p.w23[l] = addW(m23.w[l], l == 0 ? 486 : 54, 54, l == 0 ? 1 : 0);
        p.b23[l] = addB(m23.b[l], 54);
    }

    k_prep<<<nd, 256, 0, stream>>>(pa);
    k_norm1<<<512, 256, 0, stream>>>(p.fea, partial, p.n);
    k_norm2<<<1, 256, 0, stream>>>(partial, wsf);
    int grid = (p.n + 63) / 64;
    k_main<<<grid, 128, 0, stream>>>(p);
    k_nei<<<grid, 128, 0, stream>>>(p);
}